// Spatial_Attention2_38465727103731
// MI455X (gfx1250) — compile-verified
//
#include <hip/hip_runtime.h>
#include <hip/hip_bf16.h>

// Problem constants
#define BTT 96      // B*T
#define NN  1024    // nodes
#define DD  64      // head dim

typedef __attribute__((ext_vector_type(2))) float v2f;
typedef __attribute__((ext_vector_type(8))) float v8f;
typedef __attribute__((ext_vector_type(4))) unsigned int v4u;
typedef __attribute__((ext_vector_type(8))) int v8i;
typedef __attribute__((ext_vector_type(4))) int v4i;

__device__ __forceinline__ v8f wmma4(v2f a, v2f b, v8f c) {
  // D = A(16x4 f32) * B(4x16 f32) + C(16x16 f32)
  return __builtin_amdgcn_wmma_f32_16x16x4_f32(false, a, false, b, (short)0, c,
                                               false, false);
}

// ---------------------------------------------------------------------------
// TDM: DMA a 64x64 f32 tile (row stride 64 floats in memory) into LDS,
// padding each 64-DWORD row with 8 DWORDs => LDS row stride 72 floats
// (72*2 mod 64 == 16 banks, so the two 16-lane halves of a B-operand
//  ds_load hit disjoint bank ranges).
// D# layout per CDNA5 ISA.
// This toolchain exposes the 6-arg builtin:
//   (uint32x4 g0, int32x8 g1, int32x4, int32x4, int32x8, i32 cpol)
// ---------------------------------------------------------------------------
#define KCH   64                      // rows per chunk
#define LDSW  72                      // padded LDS row stride (floats)

__device__ __forceinline__ void tdm_load_tile(unsigned lds_byte_off,
                                              const float* gsrc) {
  unsigned long long ga = (unsigned long long)gsrc;
  v4u g0;
  g0[0] = 1u;                                   // count=1, user descriptor
  g0[1] = lds_byte_off;                         // LDS dest address (bytes)
  g0[2] = (unsigned)(ga & 0xFFFFFFFFull);       // global_addr[31:0]
  g0[3] = (unsigned)((ga >> 32) & 0x01FFFFFFull) | (2u << 30);  // [56:32]|type=2
  v8i g1;
  // data_size=4B(code2), pad_enable, pad_interval=64DW(code5), pad_amount=8DW(code7)
  g1[0] = (int)0x0F520000;
  g1[1] = (int)(64u << 16);     // tensor_dim0[15:0]=64  (bits 63:48)
  g1[2] = (int)(1024u << 16);   // tensor_dim0 hi=0 | tensor_dim1[15:0]=1024
  g1[3] = (int)(64u << 16);     // tensor_dim1 hi=0 | tile_dim0=64
  g1[4] = (int)KCH;             // tile_dim1=64, tile_dim2=0
  g1[5] = (int)64;              // tensor_dim0_stride[31:0]=64
  g1[6] = 0;                    // stride hi / dim1_stride lo
  g1[7] = 0;
  v4i z4 = {0, 0, 0, 0};
  v8i z8 = {0, 0, 0, 0, 0, 0, 0, 0};
  __builtin_amdgcn_tensor_load_to_lds(g0, g1, z4, z4, z8, 0);
}

// ---------------------------------------------------------------------------
// Kernel 1: QKV projection.  X[98304,64] @ W[64,192] + b -> split Q/K/V.
// One wave computes one 16x16 output tile.  6144 m-tiles x 12 n-tiles.
// ---------------------------------------------------------------------------
__global__ __launch_bounds__(128) void qkv_kernel(
    const float* __restrict__ X, const float* __restrict__ W,
    const float* __restrict__ bias, float* __restrict__ q,
    float* __restrict__ k, float* __restrict__ v) {
  const int wave = threadIdx.x >> 5;
  const int lane = threadIdx.x & 31;
  const int tile = blockIdx.x * 4 + wave;
  const int mtile = tile / 12;
  const int ntile = tile % 12;
  if (mtile >= 6144) return;
  const int lh = lane >> 4;   // lane-half
  const int lc = lane & 15;   // lane-in-half
  const int m = mtile * 16 + lc;  // A-matrix row for this lane
  const int n0 = ntile * 16;

  v8f c0 = {}, c1 = {};
#pragma unroll
  for (int k0 = 0; k0 < 64; k0 += 8) {
    const int ka = k0 + 2 * lh;
    v2f a0 = *(const v2f*)(X + (size_t)m * DD + ka);
    v2f a1 = *(const v2f*)(X + (size_t)m * DD + ka + 4);
    v2f b0, b1;
    b0.x = W[(size_t)(ka + 0) * 192 + n0 + lc];
    b0.y = W[(size_t)(ka + 1) * 192 + n0 + lc];
    b1.x = W[(size_t)(ka + 4) * 192 + n0 + lc];
    b1.y = W[(size_t)(ka + 5) * 192 + n0 + lc];
    c0 = wmma4(a0, b0, c0);
    c1 = wmma4(a1, b1, c1);
  }
  const float bb = bias[n0 + lc];
  const int grp = n0 >> 6;            // 0 -> Q, 1 -> K, 2 -> V
  const int dcol = (n0 & 63) + lc;    // column inside the 64-wide slice
  float* outp = (grp == 0) ? q : (grp == 1) ? k : v;
  const int rowbase = mtile * 16 + 8 * lh;
#pragma unroll
  for (int r = 0; r < 8; ++r)
    outp[(size_t)(rowbase + r) * DD + dcol] = c0[r] + c1[r] + bb;
}

// ---------------------------------------------------------------------------
// Kernel 2: P = G^T @ K per (b,t).  P[n,d] = sum_m G[m,n] * K[m,d].
// Block = 4 waves = 4 n-tiles; each wave computes a 16x64 strip (4 acc tiles)
// so G loads amortize 4x.  K_bt is staged into LDS by the Tensor Data Mover,
// double-buffered (tensor load of chunk c+1 overlaps compute on chunk c).
// ---------------------------------------------------------------------------
__global__ __launch_bounds__(128) void pmat_kernel(
    const float* __restrict__ G, const float* __restrict__ Kin,
    float* __restrict__ P) {
  const int wave = threadIdx.x >> 5;
  const int lane = threadIdx.x & 31;
  const int bt = blockIdx.x >> 4;          // 96
  const int ngrp = blockIdx.x & 15;        // 16 groups of 4 n-tiles
  const int ntile = ngrp * 4 + wave;
  const int lh = lane >> 4;
  const int lc = lane & 15;

  __shared__ float kbuf[2][KCH * LDSW];    // 2 x 18 KB

  const float* Kbt = Kin + (size_t)bt * NN * DD;
  const int n = ntile * 16 + lc;           // output row (A-matrix row)

  v8f acc[4] = {{}, {}, {}, {}};

  // Prologue: TDM chunk 0 (wave 0 only; EXEC is ignored by tensor ops)
  if (wave == 0) tdm_load_tile((unsigned)(unsigned long long)(void*)&kbuf[0][0], Kbt);

  for (int ch = 0; ch < NN / KCH; ++ch) {
    if (wave == 0) __builtin_amdgcn_s_wait_tensorcnt(0);
    __syncthreads();  // chunk ch resident for all waves
    if (ch + 1 < NN / KCH && wave == 0)
      tdm_load_tile((unsigned)(unsigned long long)(void*)&kbuf[(ch + 1) & 1][0],
                    Kbt + (size_t)(ch + 1) * KCH * DD);

    const float* kb = kbuf[ch & 1];
    const int mbase = ch * KCH;
#pragma unroll 4
    for (int ml = 0; ml < KCH; ml += 4) {
      const int ma = mbase + ml + 2 * lh;
      v2f a;
      a.x = G[(size_t)(ma + 0) * NN + n];
      a.y = G[(size_t)(ma + 1) * NN + n];
      const int lrow = (ml + 2 * lh) * LDSW;
#pragma unroll
      for (int t = 0; t < 4; ++t) {
        v2f b;
        b.x = kb[lrow + t * 16 + lc];
        b.y = kb[lrow + LDSW + t * 16 + lc];
        acc[t] = wmma4(a, b, acc[t]);
      }
    }
    __syncthreads();  // all waves done with buffer before TDM overwrites it
  }

  float* Pbt = P + (size_t)bt * NN * DD;
  const int rowbase = ntile * 16 + 8 * lh;
#pragma unroll
  for (int r = 0; r < 8; ++r)
#pragma unroll
    for (int t = 0; t < 4; ++t)
      Pbt[(size_t)(rowbase + r) * DD + t * 16 + lc] = acc[t][r];
}

// ---------------------------------------------------------------------------
// Kernel 3: flash attention with P as keys + LayerNorm epilogue.
// e = Q P^T * 0.125 ; attn = softmax(e) ; out = LN(attn @ V).
// One wave handles 16 query rows x full D=64.  Online softmax.
// ---------------------------------------------------------------------------
__global__ __launch_bounds__(128) void attn_ln_kernel(
    const float* __restrict__ Q, const float* __restrict__ P,
    const float* __restrict__ V, const float* __restrict__ gamma,
    const float* __restrict__ beta, float* __restrict__ out) {
  const int wave = threadIdx.x >> 5;
  const int lane = threadIdx.x & 31;
  const int tile = blockIdx.x * 4 + wave;  // 96 * 64 q-tiles
  const int bt = tile >> 6;
  const int qtile = tile & 63;
  const int lh = lane >> 4;
  const int lc = lane & 15;

  __shared__ float lds[4][16 * 17];  // stride 17 => conflict-free transpose
  float* myLds = lds[wave];

  const float* Qbt = Q + (size_t)bt * NN * DD;
  const float* Pbt = P + (size_t)bt * NN * DD;
  const float* Vbt = V + (size_t)bt * NN * DD;

  // Preload 16x64 Q tile as 16 A-matrix chunks (16x4 each)
  v2f qa[16];
  const int qrow = qtile * 16 + lc;
#pragma unroll
  for (int c4 = 0; c4 < 16; ++c4)
    qa[c4] = *(const v2f*)(Qbt + (size_t)qrow * DD + c4 * 4 + 2 * lh);

  v8f acc[4] = {{}, {}, {}, {}};
  float mrow[8], lrow[8];
#pragma unroll
  for (int r = 0; r < 8; ++r) { mrow[r] = -1e30f; lrow[r] = 0.0f; }

  for (int kb = 0; kb < 64; ++kb) {
    // ---- scores S = Q * P^T for this 16-key block (dual acc for ILP) ----
    v8f s0 = {}, s1 = {};
    const float* Pblk = Pbt + (size_t)(kb * 16) * DD;
#pragma unroll
    for (int c4 = 0; c4 < 16; c4 += 2) {
      v2f b0 = *(const v2f*)(Pblk + (size_t)lc * DD + c4 * 4 + 2 * lh);
      v2f b1 = *(const v2f*)(Pblk + (size_t)lc * DD + (c4 + 1) * 4 + 2 * lh);
      s0 = wmma4(qa[c4], b0, s0);
      s1 = wmma4(qa[c4 + 1], b1, s1);
    }
    v8f s;
#pragma unroll
    for (int r = 0; r < 8; ++r) s[r] = (s0[r] + s1[r]) * 0.125f;

    // ---- online softmax (rows live in a 16-lane half, one VGPR each) ----
    float pscale[8];
#pragma unroll
    for (int r = 0; r < 8; ++r) {
      float x = s[r];
#pragma unroll
      for (int off = 1; off < 16; off <<= 1) x = fmaxf(x, __shfl_xor(x, off, 32));
      const float mnew = fmaxf(mrow[r], x);
      pscale[r] = __expf(mrow[r] - mnew);
      mrow[r] = mnew;
      const float pv = __expf(s[r] - mnew);
      s[r] = pv;
      float sum = pv;
#pragma unroll
      for (int off = 1; off < 16; off <<= 1) sum += __shfl_xor(sum, off, 32);
      lrow[r] = lrow[r] * pscale[r] + sum;
    }
#pragma unroll
    for (int r = 0; r < 8; ++r) {
      acc[0][r] *= pscale[r];
      acc[1][r] *= pscale[r];
      acc[2][r] *= pscale[r];
      acc[3][r] *= pscale[r];
    }

    // ---- transpose probs (C/D layout) -> A layout through LDS ----
#pragma unroll
    for (int r = 0; r < 8; ++r) myLds[(r + 8 * lh) * 17 + lc] = s[r];
    __builtin_amdgcn_wave_barrier();  // keep DS store before DS load
    v2f pa[4];
#pragma unroll
    for (int c4 = 0; c4 < 4; ++c4) {
      pa[c4].x = myLds[lc * 17 + c4 * 4 + 2 * lh + 0];
      pa[c4].y = myLds[lc * 17 + c4 * 4 + 2 * lh + 1];
    }

    // ---- acc += probs @ V_block ----
    const float* Vblk = Vbt + (size_t)(kb * 16) * DD;
#pragma unroll
    for (int c4 = 0; c4 < 4; ++c4) {
      const int krow = c4 * 4 + 2 * lh;
#pragma unroll
      for (int t = 0; t < 4; ++t) {
        v2f b;
        b.x = Vblk[(size_t)(krow + 0) * DD + t * 16 + lc];
        b.y = Vblk[(size_t)(krow + 1) * DD + t * 16 + lc];
        acc[t] = wmma4(pa[c4], b, acc[t]);
      }
    }
  }

  // ---- epilogue: divide by row sums, LayerNorm over D=64 ----
  float invl[8];
#pragma unroll
  for (int r = 0; r < 8; ++r) invl[r] = 1.0f / lrow[r];
#pragma unroll
  for (int t = 0; t < 4; ++t)
#pragma unroll
    for (int r = 0; r < 8; ++r) acc[t][r] *= invl[r];

  float mu[8], rs[8];
#pragma unroll
  for (int r = 0; r < 8; ++r) {
    float sm = acc[0][r] + acc[1][r] + acc[2][r] + acc[3][r];
    float sq = acc[0][r] * acc[0][r] + acc[1][r] * acc[1][r] +
               acc[2][r] * acc[2][r] + acc[3][r] * acc[3][r];
#pragma unroll
    for (int off = 1; off < 16; off <<= 1) {
      sm += __shfl_xor(sm, off, 32);
      sq += __shfl_xor(sq, off, 32);
    }
    const float m = sm * (1.0f / 64.0f);
    const float var = sq * (1.0f / 64.0f) - m * m;
    mu[r] = m;
    rs[r] = rsqrtf(var + 1e-5f);
  }

  float gam[4], bet[4];
#pragma unroll
  for (int t = 0; t < 4; ++t) {
    gam[t] = gamma[t * 16 + lc];
    bet[t] = beta[t * 16 + lc];
  }

  float* obt = out + (size_t)bt * NN * DD;
  const int rowbase = qtile * 16 + 8 * lh;
#pragma unroll
  for (int r = 0; r < 8; ++r) {
    const size_t ro = (size_t)(rowbase + r) * DD;
#pragma unroll
    for (int t = 0; t < 4; ++t)
      obt[ro + t * 16 + lc] = (acc[t][r] - mu[r]) * rs[r] * gam[t] + bet[t];
  }
}

// ---------------------------------------------------------------------------
extern "C" void kernel_launch(void* const* d_in, const int* in_sizes, int n_in,
                              void* d_out, int out_size, void* d_ws,
                              size_t ws_size, hipStream_t stream) {
  const float* X     = (const float*)d_in[0];  // [96,1024,64]
  const float* G     = (const float*)d_in[1];  // [1024,1024]
  const float* W     = (const float*)d_in[2];  // [64,192]
  const float* bias  = (const float*)d_in[3];  // [192]
  const float* gamma = (const float*)d_in[4];  // [64]
  const float* beta  = (const float*)d_in[5];  // [64]
  float* out = (float*)d_out;

  const size_t per = (size_t)BTT * NN * DD;  // 6,291,456 floats per buffer
  float* q = (float*)d_ws;
  float* k = q + per;
  float* v = k + per;
  float* p = v + per;

  // 1) QKV projection: 6144 m-tiles * 12 n-tiles, 4 waves/block
  qkv_kernel<<<(6144 * 12) / 4, 128, 0, stream>>>(X, W, bias, q, k, v);
  // 2) P = G^T K : 96 bt * 16 n-groups (4 waves = 4 n-tiles each)
  pmat_kernel<<<96 * 16, 128, 0, stream>>>(G, k, p);
  // 3) fused attention (keys = P) + LayerNorm : 96 bt * 64 q-tiles
  attn_ln_kernel<<<(96 * 64) / 4, 128, 0, stream>>>(q, p, v, gamma, beta, out);
}